// LogmLayer_81149112091272
// MI455X (gfx1250) — compile-verified
//
#include <hip/hip_runtime.h>
#include <hip/hip_bf16.h>
#include <math.h>

// ---------------------------------------------------------------------------
// Batched 64x64 SPD matrix-log via 5-step QR eigen-iteration (MI455X / gfx1250)
//
//  * one matrix per workgroup, 128 threads = 4 wave32s
//  * all data resident in LDS (4 buffers of 64x65 f32 ~ 67 KB -> 4 WG / WGP)
//  * dense 64^3 matmuls (R@Q, U@Q, W@U^T) on V_WMMA_F32_16X16X4_F32,
//    A-fragments preloaded to VGPRs and reused across the 4 output tiles
//  * input copied HBM -> LDS with GLOBAL_LOAD_ASYNC_TO_LDS_B32 (ASYNCcnt),
//    overlapped with the Qt = I initialization
//  * MGS-QR on VALU, column-per-thread updates (stride 65 -> bank-conflict free)
// ---------------------------------------------------------------------------

typedef __attribute__((ext_vector_type(2))) float v2f;
typedef __attribute__((ext_vector_type(8))) float v8f;

#define LDW      65      // LDS row stride (64 + 1 pad: conflict-free columns)
#define NTHREADS 128     // 4 wave32s
#define NB       4096    // elements per 64x64 matrix

// C = A @ B (or A @ B^T when BT), all 64x64 in LDS with stride LDW.
// Wave w computes tile-row w (4 tiles of 16x16); 16x16x4 f32 WMMA, 16 K-steps.
// The 16 A fragments for this wave's tile-row are loaded once and reused for
// all 4 tile-columns (halves the DS traffic in the GEMM inner loops).
template <bool BT>
__device__ __forceinline__ void mm64(const float* __restrict__ A,
                                     const float* __restrict__ B,
                                     float* __restrict__ C,
                                     int wave, int lane) {
  const int half = lane >> 4;        // 0: K pair {0,1}, 1: K pair {2,3}
  const int lm   = lane & 15;
  const int row  = wave * 16 + lm;   // A-matrix row held by this lane

  v2f a[16];                         // all K fragments of this wave's A rows
#pragma unroll
  for (int kk = 0; kk < 16; ++kk) {
    const int ka = 4 * kk + 2 * half;
    a[kk].x = A[row * LDW + ka];
    a[kk].y = A[row * LDW + ka + 1];
  }

  for (int tc = 0; tc < 4; ++tc) {
    const int col = tc * 16 + lm;    // B/C column held by this lane
    v8f acc = {};
#pragma unroll
    for (int kk = 0; kk < 16; ++kk) {
      const int ka = 4 * kk + 2 * half;
      v2f b;
      if (BT) {                      // B^T: read B[col][k]
        b.x = B[col * LDW + ka];
        b.y = B[col * LDW + ka + 1];
      } else {                       // B: read B[k][col]
        b.x = B[ka * LDW + col];
        b.y = B[(ka + 1) * LDW + col];
      }
      acc = __builtin_amdgcn_wmma_f32_16x16x4_f32(
          /*neg_a=*/false, a[kk], /*neg_b=*/false, b,
          /*c_mod=*/(short)0, acc, /*reuse_a=*/false, /*reuse_b=*/false);
    }
#pragma unroll
    for (int r = 0; r < 8; ++r)      // D layout: vgpr r -> M = r + 8*half
      C[(wave * 16 + r + 8 * half) * LDW + col] = acc[r];
  }
}

__global__ __launch_bounds__(NTHREADS)
void logm_qr_kernel(const float* __restrict__ in, float* __restrict__ out,
                    int nbatch) {
  extern __shared__ float smem[];
  float* bufA = smem;                 // V  (A_k; destroyed by MGS)
  float* bufB = bufA + 64 * LDW;      // Q  (current orthogonal factor)
  float* bufC = bufB + 64 * LDW;      // R  (swaps with Qt)
  float* bufD = bufC + 64 * LDW;      // Qt (accumulated eigenvectors)
  float* red  = bufD + 64 * LDW;      // 72 floats reduction scratch

  const int tid  = threadIdx.x;
  const int wave = tid >> 5;
  const int lane = tid & 31;
  const int b    = blockIdx.x;
  if (b >= nbatch) return;            // uniform per block: WMMA EXEC stays ~0

  const float* gin  = in  + (size_t)b * NB;
  float*       gout = out + (size_t)b * NB;

  // ---- async DMA input HBM -> LDS (ASYNCcnt), per-lane B32 (padded rows) --
  for (int i = tid; i < NB; i += NTHREADS) {
    const int r = i >> 6, c = i & 63;
    unsigned long long gaddr = (unsigned long long)(const void*)(gin + i);
    unsigned int       laddr =
        (unsigned int)(unsigned long)(const void*)(bufA + r * LDW + c);
    asm volatile("global_load_async_to_lds_b32 %0, %1, off"
                 :: "v"(laddr), "v"(gaddr) : "memory");
  }
  // overlap: init Qt = I while the DMA is in flight
  for (int i = tid; i < NB; i += NTHREADS) {
    const int r = i >> 6, c = i & 63;
    bufD[r * LDW + c] = (r == c) ? 1.0f : 0.0f;
  }
  asm volatile("s_wait_asynccnt 0" ::: "memory");
  __syncthreads();

  float* pV = bufA; float* pQ = bufB; float* pR = bufC; float* pQt = bufD;

  for (int it = 0; it < 5; ++it) {
    // ---- modified Gram-Schmidt QR: pV -> (pQ, pR); pV destroyed -----------
    for (int i = tid; i < NB; i += NTHREADS) {
      const int r = i >> 6, c = i & 63;
      pR[r * LDW + c] = 0.0f;
    }
    __syncthreads();

    for (int j = 0; j < 64; ++j) {
      // ||v_j||^2 via LDS partials + wave shuffle reduce
      if (tid < 64) { float v = pV[tid * LDW + j]; red[tid] = v * v; }
      __syncthreads();
      if (tid < 32) {
        float s = red[tid] + red[tid + 32];
        for (int off = 16; off > 0; off >>= 1) s += __shfl_down(s, off);
        if (tid == 0) {
          float r = sqrtf(fmaxf(s, 1e-30f));
          pR[j * LDW + j] = r;
          red[64] = 1.0f / r;
        }
      }
      __syncthreads();
      if (tid < 64) pQ[tid * LDW + j] = pV[tid * LDW + j] * red[64];
      __syncthreads();
      // rank-1 update of remaining columns, one column per thread
      for (int k = j + 1 + tid; k < 64; k += NTHREADS) {
        float dot = 0.0f;
#pragma unroll 8
        for (int i = 0; i < 64; ++i) dot += pQ[i * LDW + j] * pV[i * LDW + k];
        pR[j * LDW + k] = dot;
#pragma unroll 8
        for (int i = 0; i < 64; ++i) pV[i * LDW + k] -= dot * pQ[i * LDW + j];
      }
      __syncthreads();
    }

    // ---- A_{k+1} = R @ Q  (into pV, whose contents are dead) --------------
    mm64<false>(pR, pQ, pV, wave, lane);
    __syncthreads();
    // ---- Qt = Qt @ Q (into pR, dead) then swap roles ----------------------
    mm64<false>(pQt, pQ, pR, wave, lane);
    __syncthreads();
    float* t = pQt; pQt = pR; pR = t;
  }

  // ---- smod = log(max(diag(A_5), EPS)) ------------------------------------
  if (tid < 64) red[tid] = logf(fmaxf(pV[tid * LDW + tid], 1e-6f));
  __syncthreads();
  // ---- W = Qt * diag(smod) into pV (diag already consumed) ----------------
  for (int i = tid; i < NB; i += NTHREADS) {
    const int r = i >> 6, c = i & 63;
    pV[r * LDW + c] = pQt[r * LDW + c] * red[c];
  }
  __syncthreads();
  // ---- X = W @ Qt^T into pQ (dead) ----------------------------------------
  mm64<true>(pV, pQt, pQ, wave, lane);
  __syncthreads();

  // ---- coalesced store ----------------------------------------------------
  for (int i = tid; i < NB; i += NTHREADS) {
    const int r = i >> 6, c = i & 63;
    gout[i] = pQ[r * LDW + c];
  }
}

extern "C" void kernel_launch(void* const* d_in, const int* in_sizes, int n_in,
                              void* d_out, int out_size, void* d_ws, size_t ws_size,
                              hipStream_t stream) {
  (void)n_in; (void)d_ws; (void)ws_size; (void)out_size;
  const float* in = (const float*)d_in[0];
  float* out = (float*)d_out;
  const int nbatch = in_sizes[0] / NB;                 // 4096
  const size_t smem = (size_t)(4 * 64 * LDW + 72) * sizeof(float);  // ~67 KB
  logm_qr_kernel<<<nbatch, NTHREADS, smem, stream>>>(in, out, nbatch);
}